// InternLM3CrossAttention_15083925144099
// MI455X (gfx1250) — compile-verified
//
#include <hip/hip_runtime.h>
#include <math.h>

// ---------------------------------------------------------------------------
// Problem constants (InternLM3 decode cross-attention)
// ---------------------------------------------------------------------------
#define NB    32          // batch
#define NH    32          // query heads
#define NKVH  8           // kv heads
#define GQ    4           // NH / NKVH
#define DH    128         // head dim
#define KMAX  4096        // max kv length
#define HSZ   4096        // NH * DH
#define SCALE 0.08838834764831845f   // DH^-0.5
#define LN_THETA 13.815510557964274f // ln(1e6)

typedef float v2f __attribute__((ext_vector_type(2)));
typedef float v8f __attribute__((ext_vector_type(8)));

// D = A(16x4 f32) x B(4x16 f32) + C(16x16 f32)  -- CDNA5 V_WMMA_F32_16X16X4_F32
__device__ __forceinline__ v8f wmma_f32_k4(v2f a, v2f b, v8f c) {
  return __builtin_amdgcn_wmma_f32_16x16x4_f32(
      /*neg_a=*/false, a, /*neg_b=*/false, b,
      /*c_mod=*/(short)0, c, /*reuse_a=*/false, /*reuse_b=*/false);
}

// ---------------------------------------------------------------------------
// CDNA5 async global->LDS copy (GLOBAL_LOAD_ASYNC_TO_LDS_B128, ASYNCcnt).
// Builtin signature (from hipcc diagnostic): param0 = AS1 pointer to
// int __vector(4) (non-const), param1 = AS3 pointer, then imm offset, imm cpol.
// ---------------------------------------------------------------------------
#if defined(__has_builtin)
#if __has_builtin(__builtin_amdgcn_global_load_async_to_lds_b128) && \
    __has_builtin(__builtin_amdgcn_s_wait_asynccnt)
#define USE_ASYNC_LDS 1
#endif
#endif

typedef int vsi4 __attribute__((vector_size(16)));
typedef __attribute__((address_space(1))) vsi4 vsi4_global;
typedef __attribute__((address_space(3))) vsi4 vsi4_lds;

__device__ __forceinline__ void async_copy16(const float* g, float* l) {
#if defined(USE_ASYNC_LDS)
  __builtin_amdgcn_global_load_async_to_lds_b128(
      (vsi4_global*)g, (vsi4_lds*)l, 0, 0);
#else
  *(float4*)l = *(const float4*)g;
#endif
}

__device__ __forceinline__ void async_join() {
#if defined(USE_ASYNC_LDS)
  __builtin_amdgcn_s_wait_asynccnt(0);
#endif
}

// ---------------------------------------------------------------------------
// C[32,4096] = A[32,4096] @ W[4096,4096]^T   (row-major everywhere)
// grid = (32, 2), block = 256 (8 waves). Each wave owns a 16x16 C tile.
// A panel (16 x 256) double-buffered in LDS (async-to-LDS copy of panel p+1
// overlaps the 64 WMMAs of panel p). Padded pitch 260 -> conflict-free
// fragment reads (bank = 4*m + k, distinct across lanes).
// ---------------------------------------------------------------------------
#define KPAN 256
#define APITCH (KPAN + 4)

__global__ __launch_bounds__(256) void gemm_nt_32xK(
    const float* __restrict__ A, const float* __restrict__ W,
    float* __restrict__ C) {
  __shared__ float As[2][16 * APITCH];   // 2 x 16.6 KB

  const int tid = threadIdx.x;
  const int wv  = tid >> 5;
  const int ln  = tid & 31;
  const int row0 = blockIdx.y << 4;                 // 0 or 16
  const int col0 = ((blockIdx.x << 3) + wv) << 4;   // 16-wide column tile
  const int mn    = ln & 15;            // lane -> A row (m) and B col (n)
  const int khalf = (ln >> 4) << 1;     // lane half -> K sub-offset {0,2}

  const float* wrow = W + (size_t)(col0 + mn) * HSZ + khalf;

  // stage one K-panel (16 rows x 256 cols = 1024 float4s, 4 per thread)
  auto copy_panel = [&](int buf, int kp) {
    for (int v = tid; v < 16 * (KPAN / 4); v += 256) {
      const int r  = v >> 6;            // KPAN/4 = 64 float4s per row
      const int c4 = (v & 63) << 2;
      async_copy16(&A[(size_t)(row0 + r) * HSZ + kp + c4],
                   &As[buf][r * APITCH + c4]);
    }
  };

  copy_panel(0, 0);
  async_join();
  __syncthreads();

  constexpr int NP = HSZ / KPAN;        // 16 panels
  v8f acc = {};
  for (int p = 0; p < NP; ++p) {
    const int cur = p & 1;
    if (p + 1 < NP) copy_panel(cur ^ 1, (p + 1) * KPAN);  // overlap with compute

    __builtin_prefetch(wrow + (p + 1) * KPAN, 0, 1);      // W stream prefetch

    const float* wp = wrow + p * KPAN;
    const float* as = &As[cur][mn * APITCH + khalf];
    #pragma unroll 8
    for (int k0 = 0; k0 < KPAN; k0 += 4) {
      v2f a = *(const v2f*)&as[k0];     // A[m, k0+khalf..+1] from LDS
      v2f b = *(const v2f*)&wp[k0];     // W[n, k0+khalf..+1] from global
      acc = wmma_f32_k4(a, b, acc);
    }

    async_join();                       // panel p+1 copy complete (this wave)
    __syncthreads();                    // ... and all waves
  }

  // C tile: lane -> N=mn, VGPR j -> M = j + (ln>>4)*8
  const int mbase = (ln >> 4) << 3;
  #pragma unroll
  for (int j = 0; j < 8; ++j)
    C[(size_t)(row0 + mbase + j) * HSZ + col0 + mn] = acc[j];
}

// ---------------------------------------------------------------------------
// Decode attention, one block per (kvh, b). block = 256 (8 waves).
// Dynamic LDS: q_s[16][132] (RoPE'd q, rows>=4 zero)
//              sc [4][4100] (scores -> probs)
//              red[256]     reduction scratch
// Pass 1: WMMA M=16 keys, N=16 (g in 0..3), K-loop over d. One wave / key tile.
// Pass 2: WMMA M=16 d-values, N=16 (g), K-loop over keys. One wave / d tile.
// ---------------------------------------------------------------------------
#define QPITCH (DH + 4)     // 132
#define SPITCH (KMAX + 4)   // 4100
#define SMEM_FLOATS (16 * QPITCH + GQ * SPITCH + 256)

__global__ __launch_bounds__(256) void attn_decode(
    const float* __restrict__ qin,   // [32, 4096] pre-RoPE q
    const float* __restrict__ kc,    // [B, K, KVH, DH]
    const float* __restrict__ vc,    // [B, K, KVH, DH]
    const int* __restrict__ ctx_lens,
    const int* __restrict__ positions,
    float* __restrict__ out) {       // [32, 4096]
  extern __shared__ float smem[];
  float* q_s = smem;                       // 16 x QPITCH
  float* sc  = smem + 16 * QPITCH;         // GQ x SPITCH
  float* red = sc + GQ * SPITCH;           // 256

  const int kvh = blockIdx.x;
  const int b   = blockIdx.y;
  const int tid = threadIdx.x;
  const int wv  = tid >> 5;
  const int ln  = tid & 31;
  const int mn    = ln & 15;
  const int khalf = (ln >> 4) << 1;

  const int ctx = ctx_lens[b];
  const int pos = positions[b];

  // ---- fused RoPE load of q into LDS (rows >= GQ zeroed for N padding) ----
  for (int idx = tid; idx < 16 * DH; idx += 256) {
    const int g = idx >> 7;
    const int d = idx & (DH - 1);
    float val = 0.f;
    if (g < GQ) {
      const float* qh = qin + (size_t)b * HSZ + ((kvh << 2) + g) * DH;
      const int i = (d < 64) ? d : d - 64;
      const float ang = (float)pos * __expf(-(float)i * (LN_THETA / 64.0f));
      float sn, cn;
      __sincosf(ang, &sn, &cn);
      const float q1 = qh[i], q2 = qh[i + 64];
      val = (d < 64) ? (q1 * cn - q2 * sn) : (q2 * cn + q1 * sn);
    }
    q_s[g * QPITCH + d] = val;
  }
  __syncthreads();

  // ---- pass 1: scores = (K_tile x q^T) * SCALE, masked to -1e30 ----
  const int nTiles = (ctx + 15) >> 4;
  const float* kb = kc + (size_t)b * KMAX * (NKVH * DH) + kvh * DH;
  for (int t = wv; t < nTiles; t += 8) {
    const int key0 = t << 4;
    const float* krow = kb + (size_t)(key0 + mn) * (NKVH * DH) + khalf;
    __builtin_prefetch(krow + 8 * 16 * (NKVH * DH), 0, 1);   // next key tile
    v8f acc = {};
    #pragma unroll 8
    for (int d0 = 0; d0 < DH; d0 += 4) {
      v2f a = *(const v2f*)(krow + d0);                        // K[key, d]
      v2f q = *(const v2f*)&q_s[mn * QPITCH + d0 + khalf];     // q^T[d, g]
      acc = wmma_f32_k4(a, q, acc);
    }
    if (mn < GQ) {                       // only g-columns are real
      const int mbase = (ln >> 4) << 3;
      #pragma unroll
      for (int j = 0; j < 8; ++j) {
        const int key = key0 + mbase + j;
        sc[mn * SPITCH + key] = (key < ctx) ? acc[j] * SCALE : -1e30f;
      }
    }
  }
  __syncthreads();

  // ---- masked softmax over each of the GQ score rows ----
  const int cols = nTiles << 4;          // full written range (masked tail = 0)
  const int g64  = tid >> 6;             // 4 groups of 64 threads, one per g row
  const int t64  = tid & 63;
  float mx = -1e30f;
  for (int k2 = t64; k2 < cols; k2 += 64) mx = fmaxf(mx, sc[g64 * SPITCH + k2]);
  red[tid] = mx;
  __syncthreads();
  for (int off = 32; off > 0; off >>= 1) {
    if (t64 < off) red[tid] = fmaxf(red[tid], red[tid + off]);
    __syncthreads();
  }
  const float rowmax = red[g64 << 6];
  __syncthreads();
  float sm = 0.f;
  for (int k2 = t64; k2 < cols; k2 += 64) {
    const float e = __expf(sc[g64 * SPITCH + k2] - rowmax);
    sc[g64 * SPITCH + k2] = e;
    sm += e;
  }
  red[tid] = sm;
  __syncthreads();
  for (int off = 32; off > 0; off >>= 1) {
    if (t64 < off) red[tid] += red[tid + off];
    __syncthreads();
  }
  const float inv = 1.0f / red[g64 << 6];
  __syncthreads();
  for (int k2 = t64; k2 < cols; k2 += 64) sc[g64 * SPITCH + k2] *= inv;
  __syncthreads();

  // ---- pass 2: out^T[d, g] = V^T x P, wave wv owns d-tile [wv*16, wv*16+16) ----
  const int d0 = wv << 4;
  const float* vb = vc + (size_t)b * KMAX * (NKVH * DH) + kvh * DH;
  const int ctx4 = (ctx + 3) & ~3;       // probs beyond ctx are exactly 0
  const int nmod = mn & 3;               // clamp padded N reads in-bounds
  v8f acc = {};
  for (int key0 = 0; key0 < ctx4; key0 += 4) {
    const int keyA = key0 + khalf;
    const float* v0 = vb + (size_t)keyA * (NKVH * DH) + d0 + mn;
    v2f a;                               // V^T[d, key] : two adjacent keys
    a.x = v0[0];
    a.y = v0[NKVH * DH];
    v2f p = *(const v2f*)&sc[nmod * SPITCH + keyA];  // P[g, key..key+1]
    acc = wmma_f32_k4(a, p, acc);
  }
  if (mn < GQ) {
    float* orow = out + (size_t)b * HSZ + ((kvh << 2) + mn) * DH + d0;
    const int mbase = (ln >> 4) << 3;
    #pragma unroll
    for (int j = 0; j < 8; ++j) orow[mbase + j] = acc[j];
  }
}

// ---------------------------------------------------------------------------
// Launch: q-proj GEMM -> fused RoPE+attention -> output GEMM
// ---------------------------------------------------------------------------
extern "C" void kernel_launch(void* const* d_in, const int* in_sizes, int n_in,
                              void* d_out, int out_size, void* d_ws, size_t ws_size,
                              hipStream_t stream) {
  const float* hidden    = (const float*)d_in[0];
  const int*   positions = (const int*)d_in[1];
  const float* k_cache   = (const float*)d_in[2];
  const float* v_cache   = (const float*)d_in[3];
  const int*   ctx_lens  = (const int*)d_in[4];
  const float* wq        = (const float*)d_in[5];
  const float* wo        = (const float*)d_in[6];
  float* out = (float*)d_out;

  float* q_ws    = (float*)d_ws;            // [32, 4096]
  float* attn_ws = q_ws + NB * HSZ;         // [32, 4096]

  dim3 gemm_grid(HSZ / 16 / 8, NB / 16);    // (32, 2)
  gemm_nt_32xK<<<gemm_grid, 256, 0, stream>>>(hidden, wq, q_ws);

  attn_decode<<<dim3(NKVH, NB), 256, SMEM_FLOATS * sizeof(float), stream>>>(
      q_ws, k_cache, v_cache, ctx_lens, positions, attn_ws);

  gemm_nt_32xK<<<gemm_grid, 256, 0, stream>>>(attn_ws, wo, out);
}